// GAT_59184649339054
// MI455X (gfx1250) — compile-verified
//
#include <hip/hip_runtime.h>
#include <math.h>

#define N_NODES 50000
#define N_EDGES 1600000
#define HIN     7
#define HDIM    128
#define FCDIM   256
#define NCLS    2
#define NGRAPH  2
#define E_TOT   (N_EDGES + N_NODES)

typedef __attribute__((ext_vector_type(2))) float v2f;
typedef __attribute__((ext_vector_type(8))) float v8f;

// Hardware fp32 atomic add (GLOBAL_ATOMIC_ADD_F32) — avoids CAS-loop lowering.
__device__ __forceinline__ void fatomic_add(float* p, float v) {
    unsafeAtomicAdd(p, v);
}

// ---- monotone float<->uint encoding for atomicMax on signed floats ----
__device__ __forceinline__ unsigned f2ord(float f) {
    unsigned u = __float_as_uint(f);
    return (u & 0x80000000u) ? ~u : (u | 0x80000000u);
}
__device__ __forceinline__ float ord2f(unsigned u) {
    return (u & 0x80000000u) ? __uint_as_float(u ^ 0x80000000u) : __uint_as_float(~u);
}

// ---- per-layer init: zero acc, denom; emax := encoded(-inf); optionally pool bufs ----
__global__ void k_init(float* __restrict__ acc, unsigned* __restrict__ emaxU,
                       float* __restrict__ denom, float* __restrict__ sums,
                       float* __restrict__ cnt, int zero_pool) {
    size_t i = (size_t)blockIdx.x * blockDim.x + threadIdx.x;
    size_t nh = (size_t)N_NODES * HDIM;
    if (i < nh) acc[i] = 0.f;
    if (i < N_NODES) { emaxU[i] = 0u; denom[i] = 0.f; }
    if (zero_pool) {
        if (i < NGRAPH * HDIM) sums[i] = 0.f;
        if (i < NGRAPH) cnt[i] = 0.f;
    }
}

// ---- layer 0: [N,7] x [7,128] for both Wl and Wr (K too small for WMMA) ----
__global__ void k_gemm_l0(const float* __restrict__ x,
                          const float* __restrict__ Wl, const float* __restrict__ bl,
                          const float* __restrict__ Wr, const float* __restrict__ br,
                          float* __restrict__ xl, float* __restrict__ xr) {
    int i = blockIdx.x * blockDim.x + threadIdx.x;
    if (i >= N_NODES * HDIM) return;
    int n = i / HDIM, c = i % HDIM;
    float al = bl[c], ar = br[c];
#pragma unroll
    for (int k = 0; k < HIN; ++k) {
        float xv = x[n * HIN + k];
        al += xv * Wl[k * HDIM + c];
        ar += xv * Wr[k * HDIM + c];
    }
    xl[i] = al; xr[i] = ar;
}

// ---- layers 1/2: [N,128] x [128,128] + bias via V_WMMA_F32_16X16X4_F32 ----
// Block = 8 waves; wave (threadIdx.y) owns one 16x16 output tile of a 16-row slab.
// A frag (16x4 f32, ISA 7.12.2): lane<16 holds M=lane,K={k0,k0+1}; lane>=16 M=lane-16,K={k0+2,k0+3}.
// B frag (4x16): VGPR0 = row K=k0(+2 for hi lanes), VGPR1 = K=k0+1(+3), col = lane&15.
// C/D: lane&15 = col, rows = (lane>=16?8:0)+v across 8 VGPRs.
__global__ void __launch_bounds__(256)
k_gemm_wmma(const float* __restrict__ A, const float* __restrict__ Wm,
            const float* __restrict__ bias, float* __restrict__ out) {
    int lane  = threadIdx.x;            // 0..31
    int ct    = threadIdx.y;            // col tile 0..7
    int mrow  = blockIdx.x * 16;
    int m     = lane & 15;
    int khalf = (lane >> 4) << 1;       // 0 or 2
    int col   = ct * 16 + m;
    const float* arow = A + (size_t)(mrow + m) * HDIM;

    v8f c = {0.f, 0.f, 0.f, 0.f, 0.f, 0.f, 0.f, 0.f};
    for (int k0 = 0; k0 < HDIM; k0 += 4) {
        v2f a, b;
        a.x = arow[k0 + khalf];
        a.y = arow[k0 + khalf + 1];
        const float* bp = Wm + (size_t)(k0 + khalf) * HDIM + col;
        b.x = bp[0];
        b.y = bp[HDIM];
        c = __builtin_amdgcn_wmma_f32_16x16x4_f32(false, a, false, b, (short)0, c, false, false);
    }
    float bv   = bias[col];
    int   rb   = mrow + ((lane >> 4) << 3);
#pragma unroll
    for (int v = 0; v < 8; ++v)
        out[(size_t)(rb + v) * HDIM + col] = c[v] + bv;
}

// ---- edge pass 1: e = leaky_relu(xl[s]+xr[d]) . att ; segment max over dst ----
__global__ void __launch_bounds__(256)
k_edge_logits(const float* __restrict__ xl, const float* __restrict__ xr,
              const int* __restrict__ ei, const float* __restrict__ att,
              float* __restrict__ e, unsigned* __restrict__ emaxU) {
    int wib  = threadIdx.x >> 5;
    int lane = threadIdx.x & 31;
    long long idx = (long long)blockIdx.x * 8 + wib;
    if (idx >= E_TOT) return;
    int s, d;
    if (idx < N_EDGES) { s = ei[idx]; d = ei[N_EDGES + idx]; }
    else               { s = d = (int)(idx - N_EDGES); }
    const float4 a  = *(const float4*)(xl + (size_t)s * HDIM + lane * 4);
    const float4 b  = *(const float4*)(xr + (size_t)d * HDIM + lane * 4);
    const float4 av = *(const float4*)(att + lane * 4);
    float p = 0.f, t;
    t = a.x + b.x; t = t >= 0.f ? t : 0.2f * t; p += t * av.x;
    t = a.y + b.y; t = t >= 0.f ? t : 0.2f * t; p += t * av.y;
    t = a.z + b.z; t = t >= 0.f ? t : 0.2f * t; p += t * av.z;
    t = a.w + b.w; t = t >= 0.f ? t : 0.2f * t; p += t * av.w;
#pragma unroll
    for (int off = 16; off > 0; off >>= 1) p += __shfl_down(p, off, 32);
    if (lane == 0) {
        e[idx] = p;
        atomicMax(emaxU + d, f2ord(p));   // integer atomic: native
    }
}

// ---- edge pass 2: w = exp(e - emax[d]); denom[d] += w ----
__global__ void k_edge_softmax(const int* __restrict__ ei, float* __restrict__ ew,
                               const unsigned* __restrict__ emaxU, float* __restrict__ denom) {
    long long i = (long long)blockIdx.x * blockDim.x + threadIdx.x;
    if (i >= E_TOT) return;
    int d = (i < N_EDGES) ? ei[N_EDGES + i] : (int)(i - N_EDGES);
    float w = expf(ew[i] - ord2f(emaxU[d]));
    ew[i] = w;
    fatomic_add(denom + d, w);
}

// ---- edge pass 3: acc[d] += (w/denom[d]) * xl[s] ----
__global__ void __launch_bounds__(256)
k_edge_aggregate(const float* __restrict__ xl, const int* __restrict__ ei,
                 const float* __restrict__ ew, const float* __restrict__ denom,
                 float* __restrict__ acc) {
    int wib  = threadIdx.x >> 5;
    int lane = threadIdx.x & 31;
    long long idx = (long long)blockIdx.x * 8 + wib;
    if (idx >= E_TOT) return;
    int s, d;
    if (idx < N_EDGES) { s = ei[idx]; d = ei[N_EDGES + idx]; }
    else               { s = d = (int)(idx - N_EDGES); }
    float alpha = ew[idx] / denom[d];
    const float4 a = *(const float4*)(xl + (size_t)s * HDIM + lane * 4);
    float* ap = acc + (size_t)d * HDIM + lane * 4;
    fatomic_add(ap + 0, alpha * a.x);
    fatomic_add(ap + 1, alpha * a.y);
    fatomic_add(ap + 2, alpha * a.z);
    fatomic_add(ap + 3, alpha * a.w);
}

// ---- bias + ELU (in place); optionally accumulate mean-pool buffers ----
__global__ void k_finalize(float* __restrict__ acc, const float* __restrict__ bo,
                           const int* __restrict__ batch, float* __restrict__ sums,
                           float* __restrict__ cnt, int do_pool) {
    int i = blockIdx.x * blockDim.x + threadIdx.x;
    if (i >= N_NODES * HDIM) return;
    int n = i / HDIM, c = i % HDIM;
    float v = acc[i] + bo[c];
    v = v > 0.f ? v : (expf(v) - 1.f);
    acc[i] = v;
    if (do_pool) {
        int g = batch[n];
        fatomic_add(sums + g * HDIM + c, v);
        if (c == 0) fatomic_add(cnt + g, 1.0f);
    }
}

// ---- mean pool -> fc1+relu -> fc2 -> log_softmax (G=2: single block) ----
__global__ void __launch_bounds__(256)
k_head(const float* __restrict__ sums, const float* __restrict__ cnt,
       const float* __restrict__ fc1W, const float* __restrict__ fc1b,
       const float* __restrict__ fc2W, const float* __restrict__ fc2b,
       float* __restrict__ out) {
    __shared__ float g[NGRAPH][HDIM];
    __shared__ float z[NGRAPH][FCDIM];
    __shared__ float lg[NGRAPH][NCLS];
    int tid = threadIdx.x;
    for (int i = tid; i < NGRAPH * HDIM; i += 256) {
        int gi = i / HDIM, c = i % HDIM;
        g[gi][c] = sums[i] / fmaxf(cnt[gi], 1.0f);
    }
    __syncthreads();
    for (int i = tid; i < NGRAPH * FCDIM; i += 256) {
        int gi = i / FCDIM, f = i % FCDIM;
        float a = fc1b[f];
        for (int k = 0; k < HDIM; ++k) a += g[gi][k] * fc1W[k * FCDIM + f];
        z[gi][f] = fmaxf(a, 0.f);
    }
    __syncthreads();
    if (tid < NGRAPH * NCLS) {
        int gi = tid / NCLS, c = tid % NCLS;
        float a = fc2b[c];
        for (int k = 0; k < FCDIM; ++k) a += z[gi][k] * fc2W[k * NCLS + c];
        lg[gi][c] = a;
    }
    __syncthreads();
    if (tid < NGRAPH) {
        float m = fmaxf(lg[tid][0], lg[tid][1]);
        float lse = m + logf(expf(lg[tid][0] - m) + expf(lg[tid][1] - m));
        out[tid * NCLS + 0] = lg[tid][0] - lse;
        out[tid * NCLS + 1] = lg[tid][1] - lse;
    }
}

extern "C" void kernel_launch(void* const* d_in, const int* in_sizes, int n_in,
                              void* d_out, int out_size, void* d_ws, size_t ws_size,
                              hipStream_t stream) {
    const float* x     = (const float*)d_in[0];
    const int*   ei    = (const int*)d_in[1];
    const int*   batch = (const int*)d_in[2];
    const float* Wl[3] = {(const float*)d_in[3],  (const float*)d_in[9],  (const float*)d_in[15]};
    const float* bl[3] = {(const float*)d_in[4],  (const float*)d_in[10], (const float*)d_in[16]};
    const float* Wr[3] = {(const float*)d_in[5],  (const float*)d_in[11], (const float*)d_in[17]};
    const float* br[3] = {(const float*)d_in[6],  (const float*)d_in[12], (const float*)d_in[18]};
    const float* att[3]= {(const float*)d_in[7],  (const float*)d_in[13], (const float*)d_in[19]};
    const float* bo[3] = {(const float*)d_in[8],  (const float*)d_in[14], (const float*)d_in[20]};
    const float* fc1W  = (const float*)d_in[21];
    const float* fc1b  = (const float*)d_in[22];
    const float* fc2W  = (const float*)d_in[23];
    const float* fc2b  = (const float*)d_in[24];

    float* ws = (float*)d_ws;
    size_t NH = (size_t)N_NODES * HDIM;
    float*    h     = ws;                  // node features / aggregation accumulator
    float*    xl    = ws + NH;
    float*    xr    = ws + 2 * NH;
    float*    ew    = ws + 3 * NH;         // per-edge logit -> weight
    unsigned* emaxU = (unsigned*)(ws + 3 * NH + (size_t)E_TOT);
    float*    denom = ws + 3 * NH + (size_t)E_TOT + N_NODES;
    float*    sums  = denom + N_NODES;
    float*    cnt   = sums + NGRAPH * HDIM;

    int nh_blocks     = (int)((NH + 255) / 256);
    int e_wave_blocks = (E_TOT + 7) / 8;
    int e_thr_blocks  = (E_TOT + 255) / 256;

    for (int L = 0; L < 3; ++L) {
        if (L == 0) {
            k_gemm_l0<<<nh_blocks, 256, 0, stream>>>(x, Wl[0], bl[0], Wr[0], br[0], xl, xr);
        } else {
            dim3 blk(32, 8);
            k_gemm_wmma<<<N_NODES / 16, blk, 0, stream>>>(h, Wl[L], bl[L], xl);
            k_gemm_wmma<<<N_NODES / 16, blk, 0, stream>>>(h, Wr[L], br[L], xr);
        }
        k_init<<<nh_blocks, 256, 0, stream>>>(h, emaxU, denom, sums, cnt, L == 2 ? 1 : 0);
        k_edge_logits<<<e_wave_blocks, 256, 0, stream>>>(xl, xr, ei, att[L], ew, emaxU);
        k_edge_softmax<<<e_thr_blocks, 256, 0, stream>>>(ei, ew, emaxU, denom);
        k_edge_aggregate<<<e_wave_blocks, 256, 0, stream>>>(xl, ei, ew, denom, h);
        k_finalize<<<nh_blocks, 256, 0, stream>>>(h, bo[L], batch, sums, cnt, L == 2 ? 1 : 0);
    }
    k_head<<<1, 256, 0, stream>>>(sums, cnt, fc1W, fc1b, fc2W, fc2b, (float*)d_out);
}